// low_rank_rnn_11862699671566
// MI455X (gfx1250) — compile-verified
//
#include <hip/hip_runtime.h>
#include <math.h>

// erf slope: sqrt(pi)/2
#define ALPHA_ERF 0.8862269254527580137f

typedef __attribute__((ext_vector_type(2))) float v2f;
typedef __attribute__((ext_vector_type(8))) float v8f;

// ---------------------------------------------------------------------------
// Pass 1: per-column partial reductions of phi = erf(alpha*h) against
//         w_r = v[:,r] (r=0..3) and w_4 = z.
// partial[(chunk*5 + r)*B + b] = sum_{n in chunk} w_r[n] * erf(alpha*h[n,b])
// Each thread owns 2 consecutive columns (float2 loads -> 256B per wave32).
// ---------------------------------------------------------------------------
__global__ __launch_bounds__(256) void k1_reduce(
    const float* __restrict__ h, const float* __restrict__ v,
    const float* __restrict__ z, float* __restrict__ partial,
    int B, int nchunk)
{
    const int b0    = (blockIdx.x * 256 + threadIdx.x) * 2;
    const int chunk = blockIdx.y;
    const int nBeg  = chunk * nchunk;
    const int nEnd  = nBeg + nchunk;

    float acc[5][2];
#pragma unroll
    for (int r = 0; r < 5; ++r) { acc[r][0] = 0.0f; acc[r][1] = 0.0f; }

    for (int n = nBeg; n < nEnd; ++n) {
        const size_t hoff = (size_t)n * (size_t)B + (size_t)b0;
        // stream-ahead hint for upcoming rows of h
        if (n + 4 < nEnd) __builtin_prefetch(h + hoff + 4 * (size_t)B, 0, 1);
        const float2 hv = *(const float2*)(h + hoff);
        const float p0 = erff(ALPHA_ERF * hv.x);
        const float p1 = erff(ALPHA_ERF * hv.y);
        // wave-uniform indices -> scalar loads
        const float4 vv = *(const float4*)(v + (size_t)n * 4);
        const float  zv = z[n];
        acc[0][0] = fmaf(vv.x, p0, acc[0][0]); acc[0][1] = fmaf(vv.x, p1, acc[0][1]);
        acc[1][0] = fmaf(vv.y, p0, acc[1][0]); acc[1][1] = fmaf(vv.y, p1, acc[1][1]);
        acc[2][0] = fmaf(vv.z, p0, acc[2][0]); acc[2][1] = fmaf(vv.z, p1, acc[2][1]);
        acc[3][0] = fmaf(vv.w, p0, acc[3][0]); acc[3][1] = fmaf(vv.w, p1, acc[3][1]);
        acc[4][0] = fmaf(zv,   p0, acc[4][0]); acc[4][1] = fmaf(zv,   p1, acc[4][1]);
    }

#pragma unroll
    for (int r = 0; r < 5; ++r) {
        float2 o; o.x = acc[r][0]; o.y = acc[r][1];
        *(float2*)(partial + (size_t)(chunk * 5 + r) * (size_t)B + b0) = o;
    }
}

// ---------------------------------------------------------------------------
// Pass 2: fold chunk partials. r<4 -> raw t sums into tbuf[r*B+b];
//         r==4 -> y[b] = sum/N written directly to d_out.
// ---------------------------------------------------------------------------
__global__ __launch_bounds__(256) void k2_finalize(
    const float* __restrict__ partial, float* __restrict__ tbuf,
    float* __restrict__ y_out, int B, int nchunks, float invN)
{
    const int idx = blockIdx.x * 256 + threadIdx.x;
    if (idx >= 5 * B) return;
    const int r = idx / B;
    const int b = idx - r * B;
    float s = 0.0f;
    for (int c = 0; c < nchunks; ++c)
        s += partial[(size_t)(c * 5 + r) * (size_t)B + b];
    if (r == 4) y_out[b] = s * invN;
    else        tbuf[(size_t)r * (size_t)B + b] = s;
}

// ---------------------------------------------------------------------------
// Pass 3: h_new = h + ((u@t)/N + m*x - h), rank-4 expansion done with
//         V_WMMA_F32_16X16X4_F32 (16x16 f32 out, K=4).
// One wave handles a 16-row tile x 64 columns (4 WMMAs), reusing its A (u)
// and m registers across the column tiles.  All h/hnew accesses within a
// tile share one base address; rows map to 8KB immediate offsets.
// ---------------------------------------------------------------------------
__global__ __launch_bounds__(256) void k3_update(
    const float* __restrict__ h, const float* __restrict__ u,
    const float* __restrict__ m, const float* __restrict__ x,
    const float* __restrict__ tbuf, float* __restrict__ hnew,
    int B, float invN)
{
    const int lane = threadIdx.x & 31;
    const int wave = threadIdx.x >> 5;
    const int lo   = lane & 15;        // column within tile / row M for A
    const int hi   = lane >> 4;        // half-wave select

    const int colGroups = B >> 6;                  // groups of 64 columns
    const int gwave     = blockIdx.x * 8 + wave;
    const int rowTile   = gwave / colGroups;
    const int colGroup  = gwave - rowTile * colGroups;
    const int n0        = rowTile * 16;

    // A-matrix (u rows n0..n0+15, K=0..3).  ISA layout:
    // VGPR0: K=0 (lanes 0-15), K=2 (lanes 16-31); VGPR1: K=1, K=3.
    // Row-major u[N,4] makes this a contiguous float2 load.
    v2f a;
    {
        const float2 av = *(const float2*)(u + (size_t)(n0 + lo) * 4 + hi * 2);
        a.x = av.x; a.y = av.y;
    }

    // m values for this lane's 8 output rows (D VGPR g -> row n0 + 8*hi + g)
    float mv[8];
    {
        const float4 m0 = *(const float4*)(m + n0 + 8 * hi);
        const float4 m1 = *(const float4*)(m + n0 + 8 * hi + 4);
        mv[0] = m0.x; mv[1] = m0.y; mv[2] = m0.z; mv[3] = m0.w;
        mv[4] = m1.x; mv[5] = m1.y; mv[6] = m1.z; mv[7] = m1.w;
    }

    for (int bt = 0; bt < 4; ++bt) {
        const int col = colGroup * 64 + bt * 16 + lo;

        // B-matrix (t, 4x16): VGPR j holds row K = j + 2*hi, col = lo.
        v2f bm;
        bm.x = tbuf[(size_t)(0 + 2 * hi) * (size_t)B + col];
        bm.y = tbuf[(size_t)(1 + 2 * hi) * (size_t)B + col];

        const float xv = x[col];

        // One base offset; rows g differ by B floats -> immediate offsets.
        const size_t base = (size_t)(n0 + 8 * hi) * (size_t)B + (size_t)col;
        const float* __restrict__ hp = h + base;

        // Batched loads: single clause + one wait instead of 8 round trips.
        float hv[8];
#pragma unroll
        for (int g = 0; g < 8; ++g) hv[g] = hp[(size_t)g * B];

        v8f c = {};
        v8f d = __builtin_amdgcn_wmma_f32_16x16x4_f32(
            /*neg_a=*/false, a, /*neg_b=*/false, bm,
            /*c_mod=*/(short)0, c, /*reuse_a=*/false, /*reuse_b=*/false);

        float out[8];
#pragma unroll
        for (int g = 0; g < 8; ++g) {
            const float upd = fmaf(mv[g], xv, d[g] * invN); // wph + m*x
            out[g] = hv[g] + (upd - hv[g]);                 // DT/TAU == 1
        }

        float* __restrict__ op = hnew + base;
#pragma unroll
        for (int g = 0; g < 8; ++g)
            __builtin_nontemporal_store(out[g], op + (size_t)g * B);
    }
}

// ---------------------------------------------------------------------------
extern "C" void kernel_launch(void* const* d_in, const int* in_sizes, int n_in,
                              void* d_out, int out_size, void* d_ws, size_t ws_size,
                              hipStream_t stream) {
    const float* x = (const float*)d_in[0];  // [1,B]
    const float* h = (const float*)d_in[1];  // [N,B]
    const float* m = (const float*)d_in[2];  // [N,1]
    const float* u = (const float*)d_in[3];  // [N,4]
    const float* v = (const float*)d_in[4];  // [N,4]
    const float* z = (const float*)d_in[5];  // [N,1]

    const int B = in_sizes[0];               // 2048
    const int N = in_sizes[2];               // 4096
    const float invN = 1.0f / (float)N;

    float* y_out = (float*)d_out;            // [B]
    float* hnew  = (float*)d_out + B;        // [N,B]

    const int NCHUNKS = 32;
    const int nchunk  = N / NCHUNKS;         // 128

    float* partial = (float*)d_ws;                              // 32*5*B floats
    float* tbuf    = partial + (size_t)NCHUNKS * 5 * (size_t)B; // 4*B floats

    // Pass 1: 256 threads x 2 cols = 512 cols/block
    dim3 g1(B / 512, NCHUNKS);
    k1_reduce<<<g1, 256, 0, stream>>>(h, v, z, partial, B, nchunk);

    // Pass 2
    const int tot2 = 5 * B;
    k2_finalize<<<(tot2 + 255) / 256, 256, 0, stream>>>(partial, tbuf, y_out, B,
                                                        NCHUNKS, invN);

    // Pass 3: one wave per (16-row x 64-col) strip
    const int waves = (N / 16) * (B / 64);   // 8192
    k3_update<<<waves / 8, 256, 0, stream>>>(h, u, m, x, tbuf, hnew, B, invN);
}